// DropEdge_43628277793359
// MI455X (gfx1250) — compile-verified
//
#include <hip/hip_runtime.h>

#define N_NODES 50000
#define N_EDGES 800000
#define IN_F    256
#define HID     128
#define OUT_F   64

typedef __attribute__((ext_vector_type(2))) float v2f;
typedef __attribute__((ext_vector_type(8))) float v8f;

// ---------------------------------------------------------------------------
// D = A(MxK) x B(KxN) in f32 via V_WMMA_F32_16X16X4_F32.
// - 2-D register blocking: each wave owns (BM*16) x (BN*16) outputs.
// - K, N are template constants so all fragment offsets fold into the 24-bit
//   instruction offset and loads use the saddr+voffset form (32-bit indices).
// - Explicit double-buffering: fragments for step k+1 are fetched before the
//   BM*BN WMMAs of step k execute, giving each load a full MMA block of
//   latency slack instead of s_wait_loadcnt 0 right before the consumer.
//
// Caller guarantees gridDim.x * (blockDim.x/32) == (M/(16*BM)) * (N/(16*BN))
// exactly, so EXEC is all-1s at every WMMA (ISA 7.12 requirement).
//
// VGPR layouts (ISA 7.12.2, 32-bit):
//   A 16x4 : lanes 0-15 -> M=lane, K={0,1} in v[0],v[1]; lanes 16-31 -> K={2,3}
//   B 4x16 : lanes 0-15 -> N=lane, K={0,1} in v[0],v[1]; lanes 16-31 -> K={2,3}
//   C 16x16: v[i], lanes 0-15 -> (M=i, N=lane); lanes 16-31 -> (M=i+8, N=lane-16)
// ---------------------------------------------------------------------------
template <int BM, int BN, int RELU_A, int K, int N>
__global__ __launch_bounds__(256)
void gemm_f32_wmma(const float* __restrict__ A,
                   const float* __restrict__ B,
                   float* __restrict__ C)
{
    const int lane = threadIdx.x & 31;
    const int wid  = blockIdx.x * (blockDim.x >> 5) + (threadIdx.x >> 5);

    constexpr int NGN = (N >> 4) / BN;      // n-groups
    const int mg  = wid / NGN;
    const int ng  = wid - mg * NGN;
    const int m0  = mg * (BM * 16);
    const int n0  = ng * (BN * 16);

    const int lo16 = lane & 15;
    const int hi   = lane >> 4;             // 0 -> K pair {0,1}, 1 -> K pair {2,3}

    // 32-bit element offsets from the scalar kernel-arg bases.
    const unsigned aoff = (unsigned)(m0 + lo16) * K + 2u * hi;
    const unsigned boff = (unsigned)(2 * hi) * N + (unsigned)(n0 + lo16);

    v8f acc[BM][BN];
#pragma unroll
    for (int i = 0; i < BM; ++i)
#pragma unroll
        for (int j = 0; j < BN; ++j)
            acc[i][j] = (v8f){0.f, 0.f, 0.f, 0.f, 0.f, 0.f, 0.f, 0.f};

    v2f ca[BM], cb[BN], na[BM], nb[BN];

    auto loadA = [&](v2f (&fr)[BM], int k0) {
#pragma unroll
        for (int i = 0; i < BM; ++i) {
            fr[i].x = A[aoff + (unsigned)(i * 16 * K) + (unsigned)k0];
            fr[i].y = A[aoff + (unsigned)(i * 16 * K) + (unsigned)k0 + 1u];
            if (RELU_A) {
                fr[i].x = fmaxf(fr[i].x, 0.f);
                fr[i].y = fmaxf(fr[i].y, 0.f);
            }
        }
    };
    auto loadB = [&](v2f (&fr)[BN], int k0) {
#pragma unroll
        for (int j = 0; j < BN; ++j) {
            fr[j].x = B[boff + (unsigned)(k0 * N) + (unsigned)(j * 16)];
            fr[j].y = B[boff + (unsigned)(k0 * N) + (unsigned)(j * 16) + (unsigned)N];
        }
    };
    auto mma = [&](v2f (&fa)[BM], v2f (&fb)[BN]) {
#pragma unroll
        for (int i = 0; i < BM; ++i)
#pragma unroll
            for (int j = 0; j < BN; ++j)
                // (neg_a, A, neg_b, B, c_mod, C, reuse_a, reuse_b)
                acc[i][j] = __builtin_amdgcn_wmma_f32_16x16x4_f32(
                                false, fa[i], false, fb[j], (short)0, acc[i][j],
                                false, false);
    };

    loadA(ca, 0);
    loadB(cb, 0);
#pragma unroll 2
    for (int k0 = 0; k0 < K - 4; k0 += 4) {
        loadA(na, k0 + 4);                  // prefetch next step
        loadB(nb, k0 + 4);
        mma(ca, cb);                        // consume current step
#pragma unroll
        for (int i = 0; i < BM; ++i) ca[i] = na[i];
#pragma unroll
        for (int j = 0; j < BN; ++j) cb[j] = nb[j];
    }
    mma(ca, cb);                            // epilogue

#pragma unroll
    for (int i = 0; i < BM; ++i)
#pragma unroll
        for (int j = 0; j < BN; ++j) {
            float* Cp = C + (size_t)(m0 + i * 16 + 8 * hi) * N
                          + (n0 + j * 16 + lo16);
#pragma unroll
            for (int r = 0; r < 8; ++r)
                Cp[(size_t)r * N] = acc[i][j][r];
        }
}

// ---------------------------------------------------------------------------
// out[n, f] = bias[f]   (accumulator seeding; F is a power of two)
// ---------------------------------------------------------------------------
__global__ __launch_bounds__(256)
void init_bias_kernel(float* __restrict__ out, const float* __restrict__ b,
                      const int total, const int fmask)
{
    const int i = blockIdx.x * blockDim.x + threadIdx.x;
    if (i < total) out[i] = b[i & fmask];
}

// ---------------------------------------------------------------------------
// Device-scope hardware f32 atomic add, no return -> STOREcnt-tracked
// global_atomic_add_f32 resolved at L2 (s_endpgm's implicit wait drains it).
// ---------------------------------------------------------------------------
__device__ __forceinline__ void atomic_add_f32_dev(float* p, float v)
{
    asm volatile("global_atomic_add_f32 %0, %1, off scope:SCOPE_DEV"
                 :
                 : "v"((unsigned long long)(uintptr_t)p), "v"(v)
                 : "memory");
}

// ---------------------------------------------------------------------------
// Weighted scatter-add: out[dst[e], f] += w[e] * H[src[e], f]
// F (>=32) threads cooperate on one edge, so edge metadata is wave-uniform:
// readfirstlane lets the compiler fetch src/dst/w through scalar loads.
// ---------------------------------------------------------------------------
template <int F>
__global__ __launch_bounds__(256)
void edge_scatter_kernel(const float* __restrict__ H,
                         const int*   __restrict__ src,
                         const int*   __restrict__ dst,
                         const float* __restrict__ w,
                         float*       __restrict__ out,
                         const int n_edges)
{
    constexpr int EPB = 256 / F;
    const int e = __builtin_amdgcn_readfirstlane(
                      blockIdx.x * EPB + (threadIdx.x / F));
    const int f = threadIdx.x & (F - 1);
    if (e >= n_edges) return;

    const int   s  = src[e];
    const int   d  = dst[e];
    const float we = w[e];

    const float v = we * H[(size_t)s * F + f];
    atomic_add_f32_dev(&out[(size_t)d * F + f], v);
}

// ---------------------------------------------------------------------------
extern "C" void kernel_launch(void* const* d_in, const int* in_sizes, int n_in,
                              void* d_out, int out_size, void* d_ws, size_t ws_size,
                              hipStream_t stream)
{
    const float* x   = (const float*)d_in[0];   // [50000, 256]
    const int*   ei  = (const int*)  d_in[1];   // [2, 800000]
    const float* ew  = (const float*)d_in[2];   // [800000]
    const float* W1  = (const float*)d_in[3];   // [256, 128]
    const float* b1  = (const float*)d_in[4];   // [128]
    const float* W2  = (const float*)d_in[5];   // [128, 64]
    const float* b2  = (const float*)d_in[6];   // [64]
    float*       out = (float*)d_out;           // [50000, 64]

    const int* src = ei;
    const int* dst = ei + N_EDGES;

    float* h1   = (float*)d_ws;                        // [50000,128]
    float* agg1 = h1   + (size_t)N_NODES * HID;        // [50000,128]
    float* h2   = agg1 + (size_t)N_NODES * HID;        // [50000, 64]

    // ---- Layer 1: h1 = x @ W1 --------------------------------------------
    // m-groups = 3125/5 = 625, n-groups = 8/2 = 4 -> 2500 waves.
    // blockDim 128 (4 waves) -> 625 blocks, exact tiling (no EXEC holes).
    gemm_f32_wmma<5, 2, 0, IN_F, HID><<<625, 128, 0, stream>>>(x, W1, h1);

    // agg1 = b1 (broadcast), then scatter-add weighted messages.
    {
        const int total = N_NODES * HID;
        init_bias_kernel<<<(total + 255) / 256, 256, 0, stream>>>(agg1, b1, total, HID - 1);
    }
    edge_scatter_kernel<HID><<<N_EDGES / 2, 256, 0, stream>>>(h1, src, dst, ew, agg1, N_EDGES);

    // ---- Layer 2: h2 = relu(agg1) @ W2 (ReLU fused into A load) ----------
    // m-groups = 625, n-groups = 4/2 = 2 -> 1250 waves.
    // blockDim 64 (2 waves) -> 625 blocks, exact tiling.
    gemm_f32_wmma<5, 2, 1, HID, OUT_F><<<625, 64, 0, stream>>>(agg1, W2, h2);

    // out = b2 (broadcast), then scatter-add weighted messages.
    {
        const int total = N_NODES * OUT_F;
        init_bias_kernel<<<(total + 255) / 256, 256, 0, stream>>>(out, b2, total, OUT_F - 1);
    }
    edge_scatter_kernel<OUT_F><<<N_EDGES / 4, 256, 0, stream>>>(h2, src, dst, ew, out, N_EDGES);
}